// CapsLayer_65489661329574
// MI455X (gfx1250) — compile-verified
//
#include <hip/hip_runtime.h>

#define CAPS_N   10
#define IN_CAPS  1152
#define DIMS     16
#define IN_DIMS  8
#define BATCH    256
#define EPS      1e-7f

typedef __attribute__((ext_vector_type(2))) float v2f;
typedef __attribute__((ext_vector_type(8))) float v8f;

#define ICHUNK  32
#define ICHUNKS (IN_CAPS / ICHUNK)   // 36

// W j-stride in floats: 1152*16*8 = 147456 (589824 B, fits 24-bit imm offset)
#define WJSTRIDE ((size_t)IN_CAPS * 16 * 8)
// uhat layout: [i][j][bt][256]  -> j-stride = 16*256 floats (16 KB, imm offset)
#define UJSTRIDE ((size_t)16 * 256)
// uhat i-stride in floats: 10*16*256
#define UISTRIDE ((size_t)CAPS_N * 16 * 256)

// ---------------------------------------------------------------------------
// Kernel A: u_hat generation via WMMA f32 16x16x4.
//   For fixed (j,i): D(16d x 16b) = W[j,i] (16x8) @ x[b-tile,i,:]^T (8x16),
//   two K=4 WMMA steps. Explicit two-phase pipeline per i: issue all 20 W
//   loads (imm offsets) first, then stream 20 WMMAs + stores as loads land.
// ws layout: tile(i,j,bt) base = ((i*10 + j)*16 + bt)*256 floats; each lane
// stores its 8 C/D floats verbatim at +lane*8 (fully coalesced 32B/lane).
// ---------------------------------------------------------------------------
__global__ __launch_bounds__(256) void caps_uhat_wmma(
    const float* __restrict__ x,    // [256][1152][8]
    const float* __restrict__ W,    // [10][1152][16][8]
    float* __restrict__ uhat)
{
  __shared__ float xs[16 * ICHUNK * 8];   // 16 KB
  const int tid   = threadIdx.x;
  const int ibase = blockIdx.x * ICHUNK;
  const int bbase = blockIdx.y * 16;

  // stage x tile: xs[bl][il][e]
  for (int idx = tid; idx < 16 * ICHUNK * 8; idx += 256) {
    int blr = idx >> 8;                  // / (ICHUNK*8)
    int r   = idx & 255;
    xs[idx] = x[(size_t)(bbase + blr) * (IN_CAPS * 8) + (size_t)ibase * 8 + r];
  }
  __syncthreads();

  const int wave = tid >> 5;
  const int lane = tid & 31;
  const int m    = lane & 15;   // A: M=d row ; B: N=b col (same lane formula)
  const int h    = lane >> 4;   // K offset half

  for (int il = wave; il < ICHUNK; il += 8) {
    const int i = ibase + il;

    // B operand: x^T, lane m=b_local, K(e) = 2*h + vgpr  (invariant over j)
    const float* xr = xs + (m * ICHUNK + il) * 8;
    v2f b0, b1;
    b0[0] = xr[2 * h];       b0[1] = xr[2 * h + 1];
    b1[0] = xr[4 + 2 * h];   b1[1] = xr[4 + 2 * h + 1];

    const float* wbase = W + ((size_t)i * 16 + m) * 8;              // j = 0
    float* dbase = uhat + (size_t)i * UISTRIDE + (size_t)blockIdx.y * 256
                        + lane * 8;                                 // j = 0

    // prefetch next il's W rows (j=0)
    if (il + 8 < ICHUNK) __builtin_prefetch(wbase + 8 * 16 * 8, 0, 1);

    // Phase 1: issue all A-operand loads (one big clause, deep MLP).
    v2f a0[CAPS_N], a1[CAPS_N];
#pragma unroll
    for (int j = 0; j < CAPS_N; ++j) {
      const float* wrow = wbase + (size_t)j * WJSTRIDE;
      a0[j][0] = wrow[2 * h];     a0[j][1] = wrow[2 * h + 1];
      a1[j][0] = wrow[4 + 2 * h]; a1[j][1] = wrow[4 + 2 * h + 1];
    }

    // Phase 2: WMMA + store stream; waits drain loadcnt incrementally.
#pragma unroll
    for (int j = 0; j < CAPS_N; ++j) {
      v8f acc = {};
      acc = __builtin_amdgcn_wmma_f32_16x16x4_f32(false, a0[j], false, b0,
                                                  (short)0, acc, false, false);
      acc = __builtin_amdgcn_wmma_f32_16x16x4_f32(false, a1[j], false, b1,
                                                  (short)0, acc, false, false);
      *reinterpret_cast<v8f*>(dbase + (size_t)j * UJSTRIDE) = acc;
    }
  }
}

// ---------------------------------------------------------------------------
// Routing kernel: one block per batch element, 256 threads = 16 i-groups x 16
// d-lanes. 3 rounds. Uses linearity: cc2 = <u, v0+v1>, so no cc storage.
// FROM_WS=true : read u_hat from L2-resident scratch ([i][j][bt] layout:
//                per-i base + j*16KB immediate offsets).
// FROM_WS=false: recompute u_hat from W and x (x staged in LDS).
// ---------------------------------------------------------------------------
template <bool FROM_WS>
__global__ __launch_bounds__(256) void caps_route(
    const float* __restrict__ uhat,
    const float* __restrict__ x,
    const float* __restrict__ W,
    float* __restrict__ out)     // [256][10][16]
{
  __shared__ float xs[FROM_WS ? 1 : IN_CAPS * 8];
  __shared__ float sred[CAPS_N * DIMS];
  __shared__ float vcur[CAPS_N * DIMS];
  __shared__ float vsum[CAPS_N * DIMS];

  const int b     = blockIdx.x;
  const int tid   = threadIdx.x;
  const int dlane = tid & 15;
  const int ig    = tid >> 4;

  if (!FROM_WS) {
    for (int idx = tid; idx < IN_CAPS * 8; idx += 256)
      xs[idx] = x[(size_t)b * (IN_CAPS * 8) + idx];
  }
  if (tid < CAPS_N * DIMS) { vcur[tid] = 0.f; vsum[tid] = 0.f; }
  __syncthreads();

  // decode verbatim WMMA tile layout: lane = 16*(d/8)+b_local, reg = d%8
  const int bt = b >> 4, bl = b & 15;
  const size_t lanepart =
      (size_t)bt * 256 + (size_t)(((dlane >> 3) << 4) + bl) * 8 + (dlane & 7);

  for (int r = 0; r < 3; ++r) {
    const float* vp = (r == 2) ? vsum : vcur;   // round0: vcur==0 -> c=0.1
    // hoist v into registers (invariant over i within a round)
    float vreg[CAPS_N];
#pragma unroll
    for (int j = 0; j < CAPS_N; ++j) vreg[j] = vp[j * 16 + dlane];

    float sacc[CAPS_N];
#pragma unroll
    for (int j = 0; j < CAPS_N; ++j) sacc[j] = 0.f;

    for (int i = ig; i < IN_CAPS; i += 16) {
      const float* ub = uhat + (size_t)i * UISTRIDE + lanepart;  // per-i base
      float ureg[CAPS_N], cc[CAPS_N];
#pragma unroll
      for (int j = 0; j < CAPS_N; ++j) {
        float u;
        if (FROM_WS) {
          u = ub[(size_t)j * UJSTRIDE];   // 16 KB imm offsets
        } else {
          const float* wr = W + (size_t)j * WJSTRIDE + ((size_t)i * 16 + dlane) * 8;
          const float* xv = xs + i * 8;
          u = 0.f;
#pragma unroll
          for (int e = 0; e < 8; ++e) u += wr[e] * xv[e];
        }
        ureg[j] = u;
        float p = u * vreg[j];
        p += __shfl_xor(p, 1, 16);
        p += __shfl_xor(p, 2, 16);
        p += __shfl_xor(p, 4, 16);
        p += __shfl_xor(p, 8, 16);
        cc[j] = p;                      // routing logit for (j,i)
      }
      // softmax over output capsules j (per-i local)
      float mx = cc[0];
#pragma unroll
      for (int j = 1; j < CAPS_N; ++j) mx = fmaxf(mx, cc[j]);
      float cj[CAPS_N], esum = 0.f;
#pragma unroll
      for (int j = 0; j < CAPS_N; ++j) { cj[j] = __expf(cc[j] - mx); esum += cj[j]; }
      float inv = 1.f / esum;
#pragma unroll
      for (int j = 0; j < CAPS_N; ++j) sacc[j] += cj[j] * inv * ureg[j];
    }

    // reduce s over the 16 i-groups
    if (tid < CAPS_N * DIMS) sred[tid] = 0.f;
    __syncthreads();
#pragma unroll
    for (int j = 0; j < CAPS_N; ++j) atomicAdd(&sred[j * 16 + dlane], sacc[j]);
    __syncthreads();

    // squash: v = s * sq/(1+sq) / sqrt(sq+eps)
    if (tid < CAPS_N * DIMS) {
      float sv = sred[tid];
      float sq = sv * sv;
      sq += __shfl_xor(sq, 1, 16);
      sq += __shfl_xor(sq, 2, 16);
      sq += __shfl_xor(sq, 4, 16);
      sq += __shfl_xor(sq, 8, 16);
      float rr = sq / ((1.f + sq) * sqrtf(sq + EPS));
      float v  = sv * rr;
      vcur[tid] = v;
      vsum[tid] += v;
    }
    __syncthreads();
  }

  if (tid < CAPS_N * DIMS) out[(size_t)b * (CAPS_N * DIMS) + tid] = vcur[tid];
}

// ---------------------------------------------------------------------------
extern "C" void kernel_launch(void* const* d_in, const int* in_sizes, int n_in,
                              void* d_out, int out_size, void* d_ws, size_t ws_size,
                              hipStream_t stream) {
  const float* x = (const float*)d_in[0];  // [256][1152][8]
  const float* W = (const float*)d_in[1];  // [10][1152][16][8]
  float* out = (float*)d_out;              // [256][10][16]

  const size_t uhat_bytes =
      (size_t)CAPS_N * IN_CAPS * DIMS * BATCH * sizeof(float);  // 188.7 MB, ~L2-resident

  if (ws_size >= uhat_bytes) {
    float* uhat = (float*)d_ws;
    dim3 grid(ICHUNKS, BATCH / 16);
    caps_uhat_wmma<<<grid, 256, 0, stream>>>(x, W, uhat);
    caps_route<true><<<BATCH, 256, 0, stream>>>(uhat, x, W, out);
  } else {
    caps_route<false><<<BATCH, 256, 0, stream>>>(nullptr, x, W, out);
  }
}